// GridSet_11905649344546
// MI455X (gfx1250) — compile-verified
//
#include <hip/hip_runtime.h>
#include <stdint.h>

// ---------------------------------------------------------------------------
// plane is stored HWC: plane[((y*1024)+x)*32 + c], 1024*1024*32 f32 = 128 MiB
// kept in d_ws so it can stay resident in the 192 MB L2 for the gather pass.
// ---------------------------------------------------------------------------
#define NCH   32
#define WFIN  1024
#define W1    512
#define W2    256
#define W3    128

// ---------------------------------------------------------------------------
// Kernel 1: fully-fused 3-level inverse Haar DWT.
//   plane(y,x,c) = 0.125*g0(y>>3,x>>3)
//                + 0.075*( s(y>>2)*g3_0 + s(x>>2)*g3_1 + s(y>>2)s(x>>2)*g3_2 )(y>>3,x>>3)
//                + 0.100*( s(y>>1)*g2_0 + s(x>>1)*g2_1 + s(y>>1)s(x>>1)*g2_2 )(y>>2,x>>2)
//                + 0.100*( s(y)   *g1_0 + s(x)   *g1_1 + s(y)s(x)      *g1_2 )(y>>1,x>>1)
//   with s(t) = (t&1) ? +1 : -1   (0.075 = 0.125*0.6, 0.1 = 0.25*0.4 = 0.5*0.2)
// Mapping: lane = channel (wave32 == 32 channels) -> 128B coalesced HWC store.
// Block = 8 waves = 4x2 pixel tile, so every Haar 2x2 quad's coefficient reuse
// is intra-block (L1). g1 (100 MB, ~read-once) uses non-temporal loads so it
// does not push the plane out of L2.
// ---------------------------------------------------------------------------
__global__ __launch_bounds__(256) void idwt_fused_kernel(
    const float* __restrict__ g0, const float* __restrict__ g1,
    const float* __restrict__ g2, const float* __restrict__ g3,
    float* __restrict__ plane)
{
    const int c  = threadIdx.x & 31;   // lane = channel
    const int wv = threadIdx.x >> 5;   // wave in block: 0..7
    const int x  = (blockIdx.x << 2) + (wv & 3);
    const int y  = (blockIdx.y << 1) + (wv >> 2);

    // ---- level-1 detail (g1, 512x512, scale 0.2) : signs from parity(y,x)
    float d1;
    {
        const int yy = y >> 1, xx = x >> 1;
        const float sy = (y & 1) ? 1.0f : -1.0f;
        const float sx = (x & 1) ? 1.0f : -1.0f;
        const float* p = g1 + ((size_t)c * 3) * (W1 * W1) + (size_t)yy * W1 + xx;
        const float a = __builtin_nontemporal_load(p);
        const float b = __builtin_nontemporal_load(p + (size_t)(W1 * W1));
        const float d = __builtin_nontemporal_load(p + (size_t)(2 * W1 * W1));
        d1 = sy * a + sx * b + (sy * sx) * d;
    }

    // ---- level-2 detail (g2, 256x256, scale 0.4) : signs from parity(y>>1,x>>1)
    float d2;
    {
        const int yy = y >> 2, xx = x >> 2;
        const float sy = ((y >> 1) & 1) ? 1.0f : -1.0f;
        const float sx = ((x >> 1) & 1) ? 1.0f : -1.0f;
        const float* p = g2 + ((size_t)c * 3) * (W2 * W2) + (size_t)yy * W2 + xx;
        d2 = sy * p[0] + sx * p[W2 * W2] + (sy * sx) * p[2 * W2 * W2];
    }

    // ---- level-3 detail (g3, 128x128, scale 0.6) + g0 : coords (y>>3,x>>3)
    float d3, v0;
    {
        const int yy = y >> 3, xx = x >> 3;
        const float sy = ((y >> 2) & 1) ? 1.0f : -1.0f;
        const float sx = ((x >> 2) & 1) ? 1.0f : -1.0f;
        const float* p = g3 + ((size_t)c * 3) * (W3 * W3) + (size_t)yy * W3 + xx;
        d3 = sy * p[0] + sx * p[W3 * W3] + (sy * sx) * p[2 * W3 * W3];
        v0 = g0[(size_t)c * (W3 * W3) + (size_t)yy * W3 + xx];
    }

    const float val = 0.125f * v0 + 0.075f * d3 + 0.1f * d2 + 0.1f * d1;

    // HWC store: 32 lanes -> 128 contiguous bytes
    plane[(((size_t)y * WFIN) + x) * NCH + c] = val;
}

// ---------------------------------------------------------------------------
// Kernel 2: bilinear grid-sample from the L2-resident HWC plane.
// 8 lanes per point (one float4 of channels each): every corner fetch is a
// single coalesced 128 B transaction; output is a coalesced 128 B store.
// The 32 points handled by a block are staged into LDS with the CDNA5 async
// global->LDS DMA path (ASYNCcnt), removing the 8x-redundant point reads.
// ---------------------------------------------------------------------------
__global__ __launch_bounds__(256) void grid_sample_kernel(
    const float* __restrict__ pts, const float* __restrict__ plane,
    float* __restrict__ out, int npts)
{
    __shared__ float spts[64];                 // 32 points * (x,y)

    const int tid = threadIdx.x;
    const size_t base = (size_t)blockIdx.x * 32;   // first point of this block

    // --- async stage of 32 points (256 B) into LDS: 16 lanes x b128 each ---
    if (tid < 16 && (base + 2 * (size_t)tid) < (size_t)npts) {
        const uint64_t gaddr = (uint64_t)(uintptr_t)(pts + (base + 2 * (size_t)tid) * 2);
        const uint32_t laddr = (uint32_t)(uintptr_t)(&spts[tid * 4]);
        asm volatile("global_load_async_to_lds_b128 %0, %1, off"
                     :: "v"(laddr), "v"(gaddr)
                     : "memory");
    }
    asm volatile("s_wait_asynccnt 0" ::: "memory");
    __syncthreads();

    const int pl = tid >> 3;                   // point within block: 0..31
    const int q  = tid & 7;                    // channel quad: 0..7
    const size_t p = base + (size_t)pl;
    if (p >= (size_t)npts) return;

    const float px = spts[pl * 2 + 0];
    const float py = spts[pl * 2 + 1];

    float fx = (px + 1.0f) * 0.5f * 1023.0f;
    float fy = (py + 1.0f) * 0.5f * 1023.0f;
    fx = fminf(fmaxf(fx, 0.0f), 1023.0f);
    fy = fminf(fmaxf(fy, 0.0f), 1023.0f);

    const float x0f = floorf(fx);
    const float y0f = floorf(fy);
    const int x0 = (int)x0f;
    const int y0 = (int)y0f;
    const int x1 = min(x0 + 1, WFIN - 1);
    const int y1 = min(y0 + 1, WFIN - 1);
    const float wx = fx - x0f;
    const float wy = fy - y0f;

    const float w00 = (1.0f - wy) * (1.0f - wx);
    const float w01 = (1.0f - wy) * wx;
    const float w10 = wy * (1.0f - wx);
    const float w11 = wy * wx;

    const size_t qb = (size_t)q * 4;
    const float4 g00 = *(const float4*)(plane + (((size_t)y0 * WFIN) + x0) * NCH + qb);
    const float4 g01 = *(const float4*)(plane + (((size_t)y0 * WFIN) + x1) * NCH + qb);
    const float4 g10 = *(const float4*)(plane + (((size_t)y1 * WFIN) + x0) * NCH + qb);
    const float4 g11 = *(const float4*)(plane + (((size_t)y1 * WFIN) + x1) * NCH + qb);

    float4 r;
    r.x = g00.x * w00 + g01.x * w01 + g10.x * w10 + g11.x * w11;
    r.y = g00.y * w00 + g01.y * w01 + g10.y * w10 + g11.y * w11;
    r.z = g00.z * w00 + g01.z * w01 + g10.z * w10 + g11.z * w11;
    r.w = g00.w * w00 + g01.w * w01 + g10.w * w10 + g11.w * w11;

    *(float4*)(out + p * NCH + qb) = r;
}

// ---------------------------------------------------------------------------
extern "C" void kernel_launch(void* const* d_in, const int* in_sizes, int n_in,
                              void* d_out, int out_size, void* d_ws, size_t ws_size,
                              hipStream_t stream) {
    const float* pts = (const float*)d_in[0];   // (2000000, 2) f32
    const float* g0  = (const float*)d_in[1];   // (1, 32, 128, 128)
    const float* g1  = (const float*)d_in[2];   // (1, 32, 3, 512, 512)
    const float* g2  = (const float*)d_in[3];   // (1, 32, 3, 256, 256)
    const float* g3  = (const float*)d_in[4];   // (1, 32, 3, 128, 128)
    float* out   = (float*)d_out;               // (npts, 32) f32
    float* plane = (float*)d_ws;                // 1024*1024*32 f32 = 128 MiB (fits L2)

    const int npts = in_sizes[0] / 2;

    // Pass 1: fused 3-level inverse Haar DWT -> HWC plane in workspace.
    dim3 grid1(WFIN / 4, WFIN / 2);
    idwt_fused_kernel<<<grid1, 256, 0, stream>>>(g0, g1, g2, g3, plane);

    // Pass 2: bilinear sampling (32 points / block, 8 lanes per point).
    const int nblocks = (npts + 31) / 32;
    grid_sample_kernel<<<nblocks, 256, 0, stream>>>(pts, plane, out, npts);
}